// HeteroSAGE_71107478552874
// MI455X (gfx1250) — compile-verified
//
#include <hip/hip_runtime.h>
#include <hip/hip_bf16.h>

typedef __attribute__((ext_vector_type(2))) float v2f;
typedef __attribute__((ext_vector_type(8))) float v8f;

#define HID 256
#define APITCH 260   // LDS row pitch for 16xK A tiles: 260 % 64 == 4 -> conflict-free
#define EPITCH 516   // LDS row pitch for 16x512 emb tile: 516 % 64 == 4

// ---------------------------------------------------------------- utilities

__global__ void zero_kernel(float* __restrict__ p, long long n) {
  long long t = (long long)blockIdx.x * blockDim.x + threadIdx.x;
  if (t < n) p[t] = 0.0f;
}

// in place: p = relu(p/3) or p/3
__global__ void finalize_kernel(float* __restrict__ p, long long n, int do_relu) {
  long long t = (long long)blockIdx.x * blockDim.x + threadIdx.x;
  if (t >= n) return;
  float v = p[t] * (1.0f / 3.0f);
  if (do_relu) v = v > 0.0f ? v : 0.0f;
  p[t] = v;
}

// agg[row][:] /= max(cnt[row],1)   (in place -> becomes "mean")
__global__ void mean_kernel(float* __restrict__ agg, const float* __restrict__ cnt,
                            long long n, int K) {
  long long t = (long long)blockIdx.x * blockDim.x + threadIdx.x;
  if (t >= n) return;
  long long row = t / K;
  float c = cnt[row];
  agg[t] = agg[t] / (c > 1.0f ? c : 1.0f);
}

// ------------------------------------------------------- edge scatter (mean)

// agg[dst[e]][:] += xsrc[src[e]][:] ; cnt[dst[e]] += 1
__global__ void scatter_kernel(const float* __restrict__ xsrc,
                               const int* __restrict__ src,
                               const int* __restrict__ dst,
                               float* __restrict__ agg, float* __restrict__ cnt,
                               int E, int K) {
  const int chunks = K >> 2;  // float4 chunks per edge
  long long t = (long long)blockIdx.x * blockDim.x + threadIdx.x;
  long long total = (long long)E * chunks;
  if (t >= total) return;
  int e = (int)(t / chunks);
  int c = (int)(t % chunks) << 2;
  int s = src[e], d = dst[e];
  const float4 v = *(const float4*)(xsrc + (size_t)s * K + c);
  float* ap = agg + (size_t)d * K + c;
  atomicAdd(ap + 0, v.x);
  atomicAdd(ap + 1, v.y);
  atomicAdd(ap + 2, v.z);
  atomicAdd(ap + 3, v.w);
  if (c == 0) atomicAdd(cnt + d, 1.0f);
}

// --------------------------------------------------------- WMMA fp32 GEMMs
//
// Fragment layouts per CDNA5 ISA 7.12.2 (m = lane&15, kg = lane>>4):
//   A 16x4 f32 : a[v] = A[m][k + v + 2*kg]          (v2f ds_load_b64 from LDS)
//   B 4x16 f32 : b[v] = W[k + v + 2*kg][m]
//   C 16x16    : c[v] = C[v + 8*kg][m]

__device__ __forceinline__ void wmma_lds_loop(const float (*__restrict__ at)[APITCH],
                                              const float* __restrict__ W,
                                              int K, int n0, int lane, v8f& c) {
  const int m = lane & 15, kg = lane >> 4;
  for (int kk = 0; kk < K; kk += 4) {
    v2f a = *(const v2f*)&at[m][kk + 2 * kg];
    v2f b;
    b[0] = W[(size_t)(kk + 2 * kg) * HID + n0 + m];
    b[1] = W[(size_t)(kk + 2 * kg + 1) * HID + n0 + m];
    c = __builtin_amdgcn_wmma_f32_16x16x4_f32(false, a, false, b, (short)0, c,
                                              false, false);
  }
}

// out[row0:+16, :256] (+)= mean @ W_l + x_dst @ W_r + bias
// grid: (Nrows/16, 256/64), block: 128 threads (4 waves, one 16x16 tile each).
// Both 16xK A tiles staged once per block into LDS with coalesced float4 loads.
__global__ void __launch_bounds__(128)
gemm_accum_kernel(const float* __restrict__ A1, const float* __restrict__ W1,
                  const float* __restrict__ A2, const float* __restrict__ W2,
                  const float* __restrict__ bias, float* __restrict__ out,
                  int K, int init) {
  __shared__ float at[2][16][APITCH];

  const int tid = threadIdx.x;
  const int row0 = blockIdx.x * 16;

  const int kq = K >> 2;                       // float4s per row
  for (int idx = tid; idx < 16 * kq; idx += 128) {
    int i = idx / kq;
    int k = (idx % kq) << 2;
    float4 v1 = *(const float4*)(A1 + (size_t)(row0 + i) * K + k);
    float4 v2 = *(const float4*)(A2 + (size_t)(row0 + i) * K + k);
    *(float4*)&at[0][i][k] = v1;
    *(float4*)&at[1][i][k] = v2;
  }
  __syncthreads();

  const int lane = tid & 31;
  const int wave = tid >> 5;
  const int n0 = blockIdx.y * 64 + wave * 16;

  v8f c = {};
  wmma_lds_loop(at[0], W1, K, n0, lane, c);   // mean  @ Wl
  wmma_lds_loop(at[1], W2, K, n0, lane, c);   // x_dst @ Wr

  const int m = lane & 15, kg = lane >> 4;
  const float bv = bias[n0 + m];
#pragma unroll
  for (int v = 0; v < 8; ++v) {
    size_t idx = (size_t)(row0 + v + 8 * kg) * HID + n0 + m;
    float val = c[v] + bv;
    if (!init) val += out[idx];
    out[idx] = val;
  }
}

// ------------------------------------------------------------- final MLP

// per block: 16 rows of B. Gather concat(xu[uid],xi[iid]) -> LDS,
// h = relu(emb @ D1W + D1b) via WMMA, out = h @ D2W + D2b (scalar).
__global__ void __launch_bounds__(128)
mlp_kernel(const float* __restrict__ xu, const float* __restrict__ xi,
           const int* __restrict__ uid, const int* __restrict__ iid,
           const float* __restrict__ D1W, const float* __restrict__ D1b,
           const float* __restrict__ D2W, const float* __restrict__ D2b,
           float* __restrict__ out) {
  __shared__ float emb[16][EPITCH];
  __shared__ float hbuf[16][APITCH];

  const int tid = threadIdx.x;
  const int b0 = blockIdx.x * 16;

  for (int idx = tid; idx < 16 * 512; idx += 128) {
    int i = idx >> 9, j = idx & 511;
    float v;
    if (j < 256) v = xu[(size_t)uid[b0 + i] * HID + j];
    else         v = xi[(size_t)iid[b0 + i] * HID + (j - 256)];
    emb[i][j] = v;
  }
  __syncthreads();

  const int lane = tid & 31, wave = tid >> 5;
  const int m = lane & 15, kg = lane >> 4;

  for (int ct = 0; ct < 4; ++ct) {          // 4 x (4 waves x 16 cols) = 256
    const int n0 = ct * 64 + wave * 16;
    v8f c = {};
    for (int kk = 0; kk < 512; kk += 4) {
      v2f a = *(const v2f*)&emb[m][kk + 2 * kg];
      v2f b;
      b[0] = D1W[(size_t)(kk + 2 * kg) * HID + n0 + m];
      b[1] = D1W[(size_t)(kk + 2 * kg + 1) * HID + n0 + m];
      c = __builtin_amdgcn_wmma_f32_16x16x4_f32(false, a, false, b, (short)0, c,
                                                false, false);
    }
    const float bv = D1b[n0 + m];
#pragma unroll
    for (int v = 0; v < 8; ++v) {
      float h = c[v] + bv;
      hbuf[v + 8 * kg][n0 + m] = h > 0.0f ? h : 0.0f;
    }
  }
  __syncthreads();

  if (tid < 64) {
    int row = tid >> 2, col = tid & 3;
    float acc = D2b[col];
    for (int k = 0; k < 256; ++k) acc += hbuf[row][k] * D2W[k * 4 + col];
    out[(size_t)(b0 + row) * 4 + col] = acc;
  }
}

// ---------------------------------------------------------------- launcher

static inline dim3 g1(long long n, int b) { return dim3((unsigned)((n + b - 1) / b)); }

extern "C" void kernel_launch(void* const* d_in, const int* in_sizes, int n_in,
                              void* d_out, int out_size, void* d_ws, size_t ws_size,
                              hipStream_t stream) {
  const float* user_emb = (const float*)d_in[0];   // [NU,128]
  const float* item_emb = (const float*)d_in[1];   // [NI,128]
  const float* Wl1 = (const float*)d_in[2];        // [6,128,256]
  const float* bl1 = (const float*)d_in[3];        // [6,256]
  const float* Wr1 = (const float*)d_in[4];        // [6,128,256]
  const float* Wl2 = (const float*)d_in[5];        // [6,256,256]
  const float* bl2 = (const float*)d_in[6];        // [6,256]
  const float* Wr2 = (const float*)d_in[7];        // [6,256,256]
  const float* D1W = (const float*)d_in[8];        // [512,256]
  const float* D1b = (const float*)d_in[9];        // [256]
  const float* D2W = (const float*)d_in[10];       // [256,4]
  const float* D2b = (const float*)d_in[11];       // [4]
  const int* esrc[3] = {(const int*)d_in[12], (const int*)d_in[14], (const int*)d_in[16]};
  const int* edst[3] = {(const int*)d_in[13], (const int*)d_in[15], (const int*)d_in[17]};
  const int* user_ids = (const int*)d_in[18];
  const int* item_ids = (const int*)d_in[19];

  const int EMB = 128;
  const int NU = in_sizes[0] / EMB;   // 100000
  const int NI = in_sizes[1] / EMB;   // 50000
  const int E  = in_sizes[12];        // 500000
  const int B  = in_sizes[18];        // 100000

  // workspace layout (floats)
  float* agg = (float*)d_ws;                         // NU*256 (max)
  float* cnt = agg + (size_t)NU * 256;               // NU
  float* xu1 = cnt + NU;                             // NU*256
  float* xi1 = xu1 + (size_t)NU * 256;               // NI*256
  float* xu2 = xi1 + (size_t)NI * 256;               // NU*256
  float* xi2 = xu2 + (size_t)NU * 256;               // NI*256

  // one relation+direction pass: scatter-mean then fused dual-GEMM accumulate
  auto pass = [&](const float* xsrc, const int* src, const int* dst, int ndst,
                  const float* xdst, const float* Wl, const float* Wr,
                  const float* bl, float* outbuf, int K, int init) {
    long long na = (long long)ndst * K;
    zero_kernel<<<g1(na, 256), 256, 0, stream>>>(agg, na);
    zero_kernel<<<g1(ndst, 256), 256, 0, stream>>>(cnt, ndst);
    long long st = (long long)E * (K >> 2);
    scatter_kernel<<<g1(st, 256), 256, 0, stream>>>(xsrc, src, dst, agg, cnt, E, K);
    mean_kernel<<<g1(na, 256), 256, 0, stream>>>(agg, cnt, na, K);
    gemm_accum_kernel<<<dim3(ndst / 16, HID / 64), 128, 0, stream>>>(
        agg, Wl, xdst, Wr, bl, outbuf, K, init);
  };

  // ---------------- layer 1 (K = 128) ----------------
  for (int r = 0; r < 3; ++r)   // user->item relations: dst = items
    pass(user_emb, esrc[r], edst[r], NI, item_emb,
         Wl1 + (size_t)r * EMB * HID, Wr1 + (size_t)r * EMB * HID,
         bl1 + (size_t)r * HID, xi1, EMB, r == 0);
  for (int r = 0; r < 3; ++r)   // reversed relations: dst = users
    pass(item_emb, edst[r], esrc[r], NU, user_emb,
         Wl1 + (size_t)(r + 3) * EMB * HID, Wr1 + (size_t)(r + 3) * EMB * HID,
         bl1 + (size_t)(r + 3) * HID, xu1, EMB, r == 0);
  finalize_kernel<<<g1((long long)NI * HID, 256), 256, 0, stream>>>(xi1, (long long)NI * HID, 1);
  finalize_kernel<<<g1((long long)NU * HID, 256), 256, 0, stream>>>(xu1, (long long)NU * HID, 1);

  // ---------------- layer 2 (K = 256) ----------------
  for (int r = 0; r < 3; ++r)
    pass(xu1, esrc[r], edst[r], NI, xi1,
         Wl2 + (size_t)r * HID * HID, Wr2 + (size_t)r * HID * HID,
         bl2 + (size_t)r * HID, xi2, HID, r == 0);
  for (int r = 0; r < 3; ++r)
    pass(xi1, edst[r], esrc[r], NU, xu1,
         Wl2 + (size_t)(r + 3) * HID * HID, Wr2 + (size_t)(r + 3) * HID * HID,
         bl2 + (size_t)(r + 3) * HID, xu2, HID, r == 0);
  finalize_kernel<<<g1((long long)NI * HID, 256), 256, 0, stream>>>(xi2, (long long)NI * HID, 0);
  finalize_kernel<<<g1((long long)NU * HID, 256), 256, 0, stream>>>(xu2, (long long)NU * HID, 0);

  // ---------------- prediction MLP ----------------
  mlp_kernel<<<dim3(B / 16), 128, 0, stream>>>(xu2, xi2, user_ids, item_ids,
                                               D1W, D1b, D2W, D2b, (float*)d_out);
}